// QCO_2d_9749575762326
// MI455X (gfx1250) — compile-verified
//
#include <hip/hip_runtime.h>

// ---------------- problem constants ----------------
#define LQ 8
#define HH 1024
#define WW 1024
#define NPIX (HH * WW)          // 1048576
#define PW 1025
#define P2 (PW * PW)            // 1050625 padded plane size
#define QOFF 192                // output header floats (3 x 64)
#define COFF (QOFF + LQ * P2)   // co_occurrence offset in d_out

// ---------------- launch constants -----------------
#define MM_BLOCKS 256
#define STA_BLOCKS 512
#define STA_WPB 4                          // waves (32 thr) per block
#define STA_WAVES (STA_BLOCKS * STA_WPB)   // 2048
#define STA_ITERS (NPIX / (STA_WAVES * 4)) // 128 iterations, K=4 each

typedef float v2f __attribute__((ext_vector_type(2)));
typedef float v4f __attribute__((ext_vector_type(4)));
typedef float v8f __attribute__((ext_vector_type(8)));

// ---- 1) per-block min/max ----
__global__ void k_minmax_part(const float* __restrict__ x,
                              float* __restrict__ bmin, float* __restrict__ bmax) {
    __shared__ float smin[256], smax[256];
    int tid = threadIdx.x;
    float mn = __FLT_MAX__, mx = -__FLT_MAX__;
    for (int i = blockIdx.x * 256 + tid; i < NPIX; i += MM_BLOCKS * 256) {
        float v = x[i];
        mn = fminf(mn, v);
        mx = fmaxf(mx, v);
    }
    smin[tid] = mn; smax[tid] = mx;
    __syncthreads();
    for (int off = 128; off > 0; off >>= 1) {
        if (tid < off) {
            smin[tid] = fminf(smin[tid], smin[tid + off]);
            smax[tid] = fmaxf(smax[tid], smax[tid + off]);
        }
        __syncthreads();
    }
    if (tid == 0) { bmin[blockIdx.x] = smin[0]; bmax[blockIdx.x] = smax[0]; }
}

// ---- 2) final min/max -> q_levels ----
__global__ void k_minmax_final(const float* __restrict__ bmin,
                               const float* __restrict__ bmax,
                               float* __restrict__ qlev) {
    __shared__ float smin[256], smax[256];
    int tid = threadIdx.x;
    smin[tid] = bmin[tid]; smax[tid] = bmax[tid];
    __syncthreads();
    for (int off = 128; off > 0; off >>= 1) {
        if (tid < off) {
            smin[tid] = fminf(smin[tid], smin[tid + off]);
            smax[tid] = fmaxf(smax[tid], smax[tid + off]);
        }
        __syncthreads();
    }
    if (tid < LQ) {
        float mn = smin[0], mx = smax[0];
        qlev[tid] = mn + (mx - mn) * ((float)tid * (1.0f / 7.0f));
    }
}

// ---- 3) quant planes (8 x 1025 x 1025, zero padded row/col) ----
__global__ void k_quant(const float* __restrict__ x,
                        const float* __restrict__ qlev,
                        float* __restrict__ q) {
    int idx = blockIdx.x * 256 + threadIdx.x;
    if (idx >= P2) return;
    int h = idx / PW, w = idx - h * PW;
    if (h >= HH || w >= WW) {
#pragma unroll
        for (int i = 0; i < LQ; ++i) q[i * P2 + idx] = 0.0f;
        return;
    }
    float xv = x[h * WW + w];
#pragma unroll
    for (int i = 0; i < LQ; ++i) {
        float d = xv - qlev[i];
        q[i * P2 + idx] = expf(-d * d * 16.0f);  // sigma = 2/8 -> 1/sigma^2 = 16
    }
}

// ---- 4) co-occurrence: 64 planes of 1024x1024 ----
// Each thread owns 4 consecutive pixels (never crosses a row: 1024 % 4 == 0),
// so every plane-store is one 16B-aligned non-temporal global_store_b128.
// 256 MB write stream with NT hint keeps the L2-resident quant planes hot.
__global__ void k_co(const float* __restrict__ q, float* __restrict__ co) {
    int t = blockIdx.x * 256 + threadIdx.x;  // < NPIX/4, exact
    int idx = t * 4;                         // first pixel of the group
    int h = idx >> 10, w = idx & (WW - 1);
    int li = h * PW + w;                     // left  index (padded plane)
    int ri = li + PW + 1;                    // right index (h+1, w+1)

    float l[LQ][4], r[LQ][4];
#pragma unroll
    for (int i = 0; i < LQ; ++i) {
#pragma unroll
        for (int k = 0; k < 4; ++k) {
            l[i][k] = q[i * P2 + li + k];
            r[i][k] = q[i * P2 + ri + k];
        }
    }
#pragma unroll
    for (int i = 0; i < LQ; ++i) {
#pragma unroll
        for (int j = 0; j < LQ; ++j) {
            v4f v;
            v.x = l[i][0] * r[j][0];
            v.y = l[i][1] * r[j][1];
            v.z = l[i][2] * r[j][2];
            v.w = l[i][3] * r[j][3];
            __builtin_nontemporal_store(v, (v4f*)&co[(i * LQ + j) * NPIX + idx]);
        }
    }
}

// ---- 5) sta = A(8 x 1M) * B(1M x 8) via V_WMMA_F32_16X16X4_F32 ----
// A layout (16x4 f32): lanes 0-15 row M=lane, VGPR0 = K0 | K2, VGPR1 = K1 | K3.
// Rows/cols >= 8 are fed zeros; loads stay unpredicated (EXEC all-ones for WMMA).
__global__ void k_sta_wmma(const float* __restrict__ q, float* __restrict__ partials) {
    int lane = threadIdx.x & 31;
    int wave = blockIdx.x * STA_WPB + (threadIdx.x >> 5);
    int m = lane & 15;
    int hi = lane >> 4;
    bool act = m < LQ;
    int mOff = act ? m * P2 : 0;  // inactive rows read plane 0 (safe), then select 0

    v8f c = {0.f, 0.f, 0.f, 0.f, 0.f, 0.f, 0.f, 0.f};
    int base = wave * (STA_ITERS * 4);
    for (int it = 0; it < STA_ITERS; ++it, base += 4) {
        // A pixels for this lane's two VGPR slots
        int pa0 = base + (hi ? 2 : 0);
        int pa1 = base + (hi ? 3 : 1);
        // B pixels (K rows striped across lane halves per VGPR)
        int pb0 = base + hi;
        int pb1 = base + 2 + hi;
        int liA0 = (pa0 >> 10) * PW + (pa0 & 1023);
        int liA1 = (pa1 >> 10) * PW + (pa1 & 1023);
        int riB0 = (pb0 >> 10) * PW + (pb0 & 1023) + PW + 1;
        int riB1 = (pb1 >> 10) * PW + (pb1 & 1023) + PW + 1;
        float va0 = q[mOff + liA0];
        float va1 = q[mOff + liA1];
        float vb0 = q[mOff + riB0];
        float vb1 = q[mOff + riB1];
        v2f a, b;
        a.x = act ? va0 : 0.0f;
        a.y = act ? va1 : 0.0f;
        b.x = act ? vb0 : 0.0f;
        b.y = act ? vb1 : 0.0f;
        // (neg_a, A, neg_b, B, c_mod, C, reuse_a, reuse_b)
        c = __builtin_amdgcn_wmma_f32_16x16x4_f32(false, a, false, b, (short)0, c,
                                                  false, false);
    }
    // C/D layout: VGPR v, lanes 0-15 -> (M=v, N=lane). We need the 8x8 corner.
    if (lane < LQ) {
#pragma unroll
        for (int v = 0; v < LQ; ++v)
            partials[wave * 64 + v * LQ + lane] = c[v];
    }
}

// ---- 6) deterministic partial reduction + normalize + header ----
__global__ void k_final(const float* __restrict__ partials,
                        const float* __restrict__ qlev,
                        float* __restrict__ out) {
    __shared__ float sh[64];
    int t = threadIdx.x;  // 64 threads
    float s = 0.0f;
    for (int wv = 0; wv < STA_WAVES; ++wv) s += partials[wv * 64 + t];
    sh[t] = s;
    __syncthreads();
    for (int off = 32; off > 0; off >>= 1) {
        if (t < off) sh[t] += sh[t + off];
        __syncthreads();
    }
    float total = sh[0];
    int i = t >> 3, j = t & 7;
    out[t]       = qlev[j];   // q_h[i][j] = q_levels[j]
    out[64 + t]  = qlev[i];   // q_w[i][j] = q_levels[i]
    out[128 + t] = s / total; // normalized sta
}

extern "C" void kernel_launch(void* const* d_in, const int* in_sizes, int n_in,
                              void* d_out, int out_size, void* d_ws, size_t ws_size,
                              hipStream_t stream) {
    const float* x = (const float*)d_in[0];
    float* out = (float*)d_out;
    float* wsf = (float*)d_ws;

    float* bmin = wsf;            // 256
    float* bmax = wsf + 256;      // 256
    float* qlev = wsf + 512;      // 8
    float* partials = wsf + 1024; // 2048 * 64 floats

    float* quant = out + QOFF;    // 8 x 1025 x 1025
    float* co    = out + COFF;    // 8 x 8 x 1024 x 1024

    k_minmax_part<<<MM_BLOCKS, 256, 0, stream>>>(x, bmin, bmax);
    k_minmax_final<<<1, 256, 0, stream>>>(bmin, bmax, qlev);
    k_quant<<<(P2 + 255) / 256, 256, 0, stream>>>(x, qlev, quant);
    k_co<<<(NPIX / 4) / 256, 256, 0, stream>>>(quant, co);
    k_sta_wmma<<<STA_BLOCKS, STA_WPB * 32, 0, stream>>>(quant, partials);
    k_final<<<1, 64, 0, stream>>>(partials, qlev, out);
}